// KGEModel_16879221473499
// MI455X (gfx1250) — compile-verified
//
#include <hip/hip_runtime.h>

typedef __attribute__((ext_vector_type(2))) float v2f;
typedef __attribute__((ext_vector_type(8))) float v8f;

#define HIDDEN 256
#define GAMMA_F 12.0f
#define WAVES_PER_BLOCK 8
#define TILE 16  // negatives per wave (one 16x16 WMMA tile)

// One wave computes 16 TransE scores: score[m] = gamma - sum_h |hr[h] - tail[m,h]|.
// The sum over H is done on the matrix pipe: D = A x ones + C with
// V_WMMA_F32_16X16X4_F32, A[m,k] = |hr - tail| abs-diffs (exact f32 math).
__global__ __launch_bounds__(256) void transe_tailbatch_wmma(
    const float* __restrict__ ent,   // [NENTITY, H]
    const float* __restrict__ rel,   // [NRELATION, H]
    const int*   __restrict__ pos,   // [B, 3]
    const int*   __restrict__ neg,   // [B, N]
    float*       __restrict__ out,   // [B, N]
    int N)
{
    __shared__ __align__(16) float hr[HIDDEN];

    const int b = blockIdx.y;
    const int t = threadIdx.x;

    // head + relation for this batch, once per block (256 threads, 256 elems)
    {
        const long long hidx = pos[b * 3 + 0];
        const long long ridx = pos[b * 3 + 1];
        hr[t] = ent[hidx * HIDDEN + t] + rel[ridx * HIDDEN + t];
    }
    __syncthreads();

    const int wave = t >> 5;
    const int lane = t & 31;
    const int n0 = (blockIdx.x * WAVES_PER_BLOCK + wave) * TILE;
    if (n0 >= N) return;  // wave-uniform: EXEC stays all-ones for WMMA

    // Lane l and lane l+16 cooperate on the same output row m = l&15.
    const int m = lane & 15;
    int nm = n0 + m;
    if (nm >= N) nm = N - 1;  // clamp reads for a partial last tile
    const long long row = neg[(long long)b * N + nm];
    const float2* __restrict__ tail2 =
        reinterpret_cast<const float2*>(ent + row * HIDDEN);
    const float2* __restrict__ hr2 = reinterpret_cast<const float2*>(hr);

    // 16x16x4 f32 A-layout: lanes 0-15 hold K={0,1}, lanes 16-31 hold K={2,3}.
    const int off2 = (lane < 16) ? 0 : 1;  // float2-granule offset within K-quad

    __builtin_prefetch(ent + row * HIDDEN, 0, 0);  // global_prefetch_b8

    const v2f bones = {1.0f, 1.0f};  // B = all-ones -> D columns = row sums
    v8f acc0 = {};
    v8f acc1 = {};

#pragma unroll 4
    for (int c = 0; c < HIDDEN / 4; ++c) {
        const int q = 2 * c + off2;  // h = 4c + 2*off2
        const float2 tv = tail2[q];  // global_load_b64, dense 8 B/lane
        const float2 rv = hr2[q];    // ds_load_b64, broadcast (2 addrs/wave)
        v2f a;
        a.x = __builtin_fabsf(rv.x - tv.x);
        a.y = __builtin_fabsf(rv.y - tv.y);
        if (c & 1)
            acc1 = __builtin_amdgcn_wmma_f32_16x16x4_f32(
                false, a, false, bones, (short)0, acc1, false, false);
        else
            acc0 = __builtin_amdgcn_wmma_f32_16x16x4_f32(
                false, a, false, bones, (short)0, acc0, false, false);
    }

    const v8f s = acc0 + acc1;

    // D layout (16x16 f32 C/D): VGPR j @ lane 0  -> (M=j,   N=0)
    //                           VGPR j @ lane 16 -> (M=8+j, N=0)
    float* __restrict__ orow = out + (long long)b * N + n0;
    if (lane == 0) {
        if (n0 + 0 < N) orow[0] = GAMMA_F - s[0];
        if (n0 + 1 < N) orow[1] = GAMMA_F - s[1];
        if (n0 + 2 < N) orow[2] = GAMMA_F - s[2];
        if (n0 + 3 < N) orow[3] = GAMMA_F - s[3];
        if (n0 + 4 < N) orow[4] = GAMMA_F - s[4];
        if (n0 + 5 < N) orow[5] = GAMMA_F - s[5];
        if (n0 + 6 < N) orow[6] = GAMMA_F - s[6];
        if (n0 + 7 < N) orow[7] = GAMMA_F - s[7];
    } else if (lane == 16) {
        if (n0 + 8 < N) orow[8] = GAMMA_F - s[0];
        if (n0 + 9 < N) orow[9] = GAMMA_F - s[1];
        if (n0 + 10 < N) orow[10] = GAMMA_F - s[2];
        if (n0 + 11 < N) orow[11] = GAMMA_F - s[3];
        if (n0 + 12 < N) orow[12] = GAMMA_F - s[4];
        if (n0 + 13 < N) orow[13] = GAMMA_F - s[5];
        if (n0 + 14 < N) orow[14] = GAMMA_F - s[6];
        if (n0 + 15 < N) orow[15] = GAMMA_F - s[7];
    }
}

extern "C" void kernel_launch(void* const* d_in, const int* in_sizes, int n_in,
                              void* d_out, int out_size, void* d_ws, size_t ws_size,
                              hipStream_t stream) {
    (void)n_in; (void)out_size; (void)d_ws; (void)ws_size;
    const float* ent = (const float*)d_in[0];
    const float* rel = (const float*)d_in[1];
    const int*   pos = (const int*)d_in[2];
    const int*   neg = (const int*)d_in[3];
    float*       out = (float*)d_out;

    const int B = in_sizes[2] / 3;          // positive_sample is [B,3]
    const int N = in_sizes[3] / B;          // negative_sample is [B,N]

    const int per_block = WAVES_PER_BLOCK * TILE;  // 128 negatives per block
    dim3 grid((N + per_block - 1) / per_block, B);
    transe_tailbatch_wmma<<<grid, 256, 0, stream>>>(ent, rel, pos, neg, out, N);
}